// TrueBranch_68470368633594
// MI455X (gfx1250) — compile-verified
//
#include <hip/hip_runtime.h>

typedef __attribute__((ext_vector_type(2))) float v2f;
typedef __attribute__((ext_vector_type(8))) float v8f;

#define B_DIM 128
#define C_DIM 4096
#define L_WIN 32

// One wave per (batch b, 16-channel tile). Wave computes the 16x16 product
//   D[m,n] = sum_l state[b, c_m, l] * w[c_n, l]
// with 8 chained V_WMMA_F32_16X16X4_F32 and extracts the diagonal
// D[m,m] = conv_out[b, c_m]. State is read exactly once (bandwidth-optimal);
// weight rows are reused across batches from L2/WGP$.
__global__ __launch_bounds__(256) void conv_step_wmma(
    const float* __restrict__ Bx,          // [B, C, 1]
    const float* __restrict__ conv_cache,  // layer 0: [B, C, L]
    const float* __restrict__ weight,      // [C, L]
    const float* __restrict__ bias,        // [C]
    const int*   __restrict__ cache_position,
    float*       __restrict__ out)         // [B, C, 1]
{
    const int lane    = threadIdx.x & 31;
    const int wave_id = blockIdx.x * (blockDim.x >> 5) + (threadIdx.x >> 5);

    const int b  = wave_id >> 8;           // 256 channel tiles per batch
    const int cb = (wave_id & 255) << 4;   // channel tile base

    const int m    = lane & 15;            // matrix row (A) / col (B) owned by lane
    const int koff = (lane >> 4) << 1;     // lanes 0-15: K={0,1}; lanes 16-31: K={2,3}
    const int ch   = cb + m;

    int cp = cache_position[0];
    cp = cp < 0 ? 0 : (cp > L_WIN - 1 ? L_WIN - 1 : cp);
    cp = __builtin_amdgcn_readfirstlane(cp);   // uniform -> SGPR

    const float  bxv  = Bx[(size_t)b * C_DIM + ch];
    const float* srow = conv_cache + ((size_t)b * C_DIM + ch) * L_WIN;
    const v2f*   wrow = reinterpret_cast<const v2f*>(weight + (size_t)ch * L_WIN);

    // ---- Phase 1: unconditional loads (clusterable, no predication) ----
    float sa[16];   // rolled sources: srow[(l0+1)&31], srow[(l0+2)&31] per chunk
    v2f   wv[8];    // weight pairs {w[l0], w[l0+1]}, 8B-aligned
#pragma unroll
    for (int kk = 0; kk < 8; ++kk) {
        const int l0 = kk * 4 + koff;
        sa[2 * kk]     = srow[(l0 + 1) & 31];
        sa[2 * kk + 1] = srow[(l0 + 2) & 31];
        wv[kk]         = wrow[l0 >> 1];          // l0 even -> exact pair
    }

    // ---- Phase 2: register-only roll/insert selects + chained WMMAs ----
    v8f acc = {};
#pragma unroll
    for (int kk = 0; kk < 8; ++kk) {
        const int l0 = kk * 4 + koff;
        // new_state[l] = (l == cp) ? Bx : old[(l+1) & 31]  (pure v_cndmask now)
        const float a0 = (l0     == cp) ? bxv : sa[2 * kk];
        const float a1 = (l0 + 1 == cp) ? bxv : sa[2 * kk + 1];
        v2f Am = {a0, a1};
        // 8 args: (neg_a, A, neg_b, B, c_mod, C, reuse_a, reuse_b)
        acc = __builtin_amdgcn_wmma_f32_16x16x4_f32(
            false, Am, false, wv[kk], (short)0, acc, false, false);
    }

    // Diagonal extraction (C/D layout: VGPR r holds M=r on lanes 0-15 and
    // M=r+8 on lanes 16-31, N = lane & 15):
    //   D[r,r]     -> VGPR r, lane r        (r = 0..7)
    //   D[r+8,r+8] -> VGPR r, lane r + 24
    const int sel = lane & 7;
    float v = acc[0];
    v = (sel == 1) ? acc[1] : v;
    v = (sel == 2) ? acc[2] : v;
    v = (sel == 3) ? acc[3] : v;
    v = (sel == 4) ? acc[4] : v;
    v = (sel == 5) ? acc[5] : v;
    v = (sel == 6) ? acc[6] : v;
    v = (sel == 7) ? acc[7] : v;

    if (lane < 8 || lane >= 24) {                 // divergence only after WMMAs
        const int mo = (lane < 8) ? lane : (lane - 16);  // 0..7 | 8..15
        out[(size_t)b * C_DIM + cb + mo] = v + bias[cb + mo];
    }
}

extern "C" void kernel_launch(void* const* d_in, const int* in_sizes, int n_in,
                              void* d_out, int out_size, void* d_ws, size_t ws_size,
                              hipStream_t stream) {
    const float* Bx             = (const float*)d_in[0];
    const float* conv_cache     = (const float*)d_in[1];  // LAYER_IDX==0 -> base
    const float* weight         = (const float*)d_in[2];
    const float* bias           = (const float*)d_in[3];
    const int*   cache_position = (const int*)d_in[4];
    float*       out            = (float*)d_out;

    // 128 batches * 256 tiles = 32768 waves; 8 waves/block -> 4096 blocks
    dim3 grid(4096), block(256);
    conv_step_wmma<<<grid, block, 0, stream>>>(Bx, conv_cache, weight, bias,
                                               cache_position, out);
}